// PDA_69827578298469
// MI455X (gfx1250) — compile-verified
//
#include <hip/hip_runtime.h>

#define HW      784      // 28*28
#define C_DIM   1024
#define CHUNK   112      // hw points per block in fused kernel (7 * 112 = 784)
#define TILE    16       // channels staged per async tile
#define NTILES  (C_DIM / TILE)   // 64
#define LDS_BYTES (2 * TILE * CHUNK * 4)   // 14336 B double buffer

// ---- CDNA5 async global->LDS copy (per-lane, tracked by ASYNCcnt) ----
__device__ __forceinline__ void async_copy_f32(const float* src, unsigned lds_byte_off) {
    // VDST = LDS byte offset (VGPR), VADDR = 64-bit global address
    asm volatile("global_load_async_to_lds_b32 %0, %1, off"
                 :: "v"(lds_byte_off), "v"(src)
                 : "memory");
}
__device__ __forceinline__ void wait_async_le_tile() {
    asm volatile("s_wait_asynccnt 16" ::: "memory");
}
__device__ __forceinline__ void wait_async_zero() {
    asm volatile("s_wait_asynccnt 0" ::: "memory");
}

// ---------------- Kernel A: thr[n,c] = mu * max_hw(relu(x * w_c)) ----------------
__global__ void __launch_bounds__(256)
PDA_thr_kernel(const float* __restrict__ x,
               const float* __restrict__ fcw,
               const float* __restrict__ muP,
               float* __restrict__ thr) {
    const int wave = (blockIdx.x * 256 + threadIdx.x) >> 5;   // one wave per (n,c)
    const int lane = threadIdx.x & 31;
    const int c    = wave & (C_DIM - 1);
    const float wc = fcw[c];            // wave-uniform -> scalar load
    const float mu = muP[0];

    const float4* row = (const float4*)(x + (size_t)wave * HW);  // 196 float4
    float m = 0.0f;                     // relu >= 0, so max(m, v*wc) == max(m, relu(v*wc))
    #pragma unroll
    for (int k = 0; k < 2; ++k) {
        int q = k * 32 + lane;
        if (q < HW / 4) {
            float4 v = row[q];
            m = fmaxf(m, v.x * wc);
            m = fmaxf(m, v.y * wc);
            m = fmaxf(m, v.z * wc);
            m = fmaxf(m, v.w * wc);
        }
    }
    #pragma unroll
    for (int mask = 16; mask >= 1; mask >>= 1)
        m = fmaxf(m, __shfl_xor(m, mask, 32));
    if (lane == 0) thr[wave] = m * mu;
}

// ------- Kernel B: fused masked channel-sum (async-LDS double buffered) + output -------
__global__ void __launch_bounds__(128)
PDA_fused_kernel(const float* __restrict__ x,
                 const float* __restrict__ fcw,
                 const float* __restrict__ thr,
                 float* __restrict__ out) {
    extern __shared__ float sm[];              // [2][TILE][CHUNK]
    const int t      = threadIdx.x;
    const int n      = blockIdx.y;
    const int hwbase = blockIdx.x * CHUNK;

    if (t < CHUNK) {
        const float* xrow = x + (size_t)n * C_DIM * HW + hwbase + t;
        const int    nb   = n * C_DIM;
        float acc = 0.0f;

        // prime buffer 0 with channel tile 0
        #pragma unroll
        for (int ct = 0; ct < TILE; ++ct)
            async_copy_f32(xrow + ct * HW, (unsigned)((ct * CHUNK + t) * 4));

        int par = 0;
        for (int tile = 0; tile < NTILES; ++tile) {
            const int cbase = tile * TILE;
            if (tile + 1 < NTILES) {
                const int nc = cbase + TILE;
                #pragma unroll
                for (int ct = 0; ct < TILE; ++ct)
                    async_copy_f32(xrow + (nc + ct) * HW,
                                   (unsigned)(((((par ^ 1) * TILE) + ct) * CHUNK + t) * 4));
                // 16 newest may be outstanding; previous tile (in-order) is complete
                wait_async_le_tile();
            } else {
                wait_async_zero();
            }
            // consume current tile from LDS (each thread reads only its own data)
            #pragma unroll
            for (int ct = 0; ct < TILE; ++ct) {
                const int   c   = cbase + ct;
                const float wc  = fcw[c];          // uniform -> scalar load
                const float th  = thr[nb + c];     // uniform -> scalar load
                const float v   = sm[(par * TILE + ct) * CHUNK + t];
                const float cam = fmaxf(v * wc, 0.0f);
                acc += (cam > th) ? 0.0f : cam;    // drop high-activation regions
            }
            par ^= 1;
        }

        // phase 2: out[n,c,hw] = x[n,c,hw] * mdc[n,hw]
        // x re-read is L2-hot; out is write-once -> non-temporal store so the
        // 205 MB output stream does not evict x from the 192 MB L2.
        float* orow = out + (size_t)n * C_DIM * HW + hwbase + t;
        #pragma unroll 4
        for (int c = 0; c < C_DIM; ++c)
            __builtin_nontemporal_store(xrow[c * HW] * acc, orow + c * HW);
    }
}

extern "C" void kernel_launch(void* const* d_in, const int* in_sizes, int n_in,
                              void* d_out, int out_size, void* d_ws, size_t ws_size,
                              hipStream_t stream) {
    const float* x   = (const float*)d_in[0];   // [64,1024,28,28]
    const float* fcw = (const float*)d_in[1];   // [1024]
    const float* mu  = (const float*)d_in[2];   // scalar
    float*       out = (float*)d_out;
    float*       thr = (float*)d_ws;            // 65536 floats = 256 KB

    // Kernel A: 65536 (n,c) waves, 8 waves per 256-thread block
    PDA_thr_kernel<<<dim3(65536 / 8), dim3(256), 0, stream>>>(x, fcw, mu, thr);

    // Kernel B: 7 hw-chunks x 64 samples, 14 KB dynamic LDS double buffer
    PDA_fused_kernel<<<dim3(7, 64), dim3(128), LDS_BYTES, stream>>>(x, fcw, thr, out);
}